// GCN_36249523978477
// MI455X (gfx1250) — compile-verified
//
#include <hip/hip_runtime.h>
#include <hip/hip_bf16.h>
#include <math.h>

#define IN_DIM  512
#define HID     8
#define OUT_DIM 40

typedef float v2f __attribute__((ext_vector_type(2)));
typedef float v8f __attribute__((ext_vector_type(8)));

__device__ __forceinline__ void fadd_atomic(float* p, float v) {
    // hardware global_atomic_add_f32 (L2 atomic unit)
    unsafeAtomicAdd(p, v);
}

// ---------------- degree / normalization ----------------
__global__ void k_deg_init(float* __restrict__ deg, int n) {
    int i = blockIdx.x * blockDim.x + threadIdx.x;
    if (i < n) deg[i] = 1.0f;                 // self loop
}

__global__ void k_deg_count(const int* __restrict__ dst, float* __restrict__ deg, int E) {
    int e = blockIdx.x * blockDim.x + threadIdx.x;
    if (e < E) fadd_atomic(&deg[dst[e]], 1.0f);
}

__global__ void k_deg_rsqrt(float* __restrict__ deg, int n) {
    int i = blockIdx.x * blockDim.x + threadIdx.x;
    if (i < n) deg[i] = rsqrtf(deg[i]);       // deg >= 1 always
}

// ---------------- layer-1 linear: h = x @ W1 via V_WMMA_F32_16X16X4_F32 ----------------
// One wave computes a 16-row x 16-col tile (only 8 cols valid, W1 zero-padded in LDS).
// A (16x4 f32): lane m=lane&15 is row M; VGPR0/1 hold K = k0+2*(lane>>4) + {0,1}.
// B (4x16 f32): lane n=lane&15 is col N; VGPR0/1 hold K = k0+2*(lane>>4) + {0,1}.
// C/D (16x16 f32): VGPR r holds M = r + 8*(lane>>4), N = lane&15.
__global__ __launch_bounds__(256) void k_gemm1_wmma(
    const float* __restrict__ x, const float* __restrict__ W1,
    float* __restrict__ hlin, int n)
{
    __shared__ float sW[IN_DIM * 16];         // W1 zero-padded 512x16 (32 KB of 320 KB/WGP)
    const int tid = threadIdx.x;
    for (int i = tid; i < IN_DIM * 16; i += 256) {
        int k = i >> 4, j = i & 15;
        sW[i] = (j < HID) ? W1[k * HID + j] : 0.0f;
    }
    __syncthreads();

    const int lane = tid & 31;
    const int wid  = tid >> 5;
    const int ntiles = (n + 15) >> 4;
    const int tile = blockIdx.x * 8 + wid;
    if (tile >= ntiles) return;               // uniform per wave: EXEC stays all-ones for WMMA

    const int row0 = tile << 4;
    const int m    = lane & 15;               // row within tile (A), col within tile (B/D)
    const int kh   = lane >> 4;               // K-half select
    int row = row0 + m;
    if (row >= n) row = n - 1;                // clamp (n is a multiple of 16 here anyway)

    const float2* __restrict__ xp =
        reinterpret_cast<const float2*>(x) + (size_t)row * (IN_DIM / 2);

    v8f c = {};
#pragma unroll 4
    for (int k0 = 0; k0 < IN_DIM; k0 += 4) {
        const int kk = k0 + 2 * kh;
        float2 av = xp[(k0 >> 1) + kh];       // coalesced 8B loads of x
        v2f a; a.x = av.x; a.y = av.y;
        v2f b; b.x = sW[kk * 16 + m]; b.y = sW[(kk + 1) * 16 + m];
        c = __builtin_amdgcn_wmma_f32_16x16x4_f32(
                /*neg_a=*/false, a, /*neg_b=*/false, b,
                /*c_mod=*/(short)0, c, /*reuse_a=*/false, /*reuse_b=*/false);
    }

    if (m < HID) {                            // only 8 valid output columns
#pragma unroll
        for (int r = 0; r < 8; ++r) {
            int mrow = row0 + kh * 8 + r;
            if (mrow < n) hlin[(size_t)mrow * HID + m] = c[r];
        }
    }
}

// ---------------- aggregation (C = HID = 8 features) ----------------
// seed with self-loop term dinv[n]^2 * h[n]  (also initializes the buffer)
__global__ void k_agg_init(const float* __restrict__ h, const float* __restrict__ dinv,
                           float* __restrict__ agg, int total)
{
    int idx = blockIdx.x * blockDim.x + threadIdx.x;
    if (idx >= total) return;
    int node = idx >> 3;                      // /HID
    float di = dinv[node];
    agg[idx] = di * di * h[idx];
}

__global__ void k_agg_edges(const int* __restrict__ src, const int* __restrict__ dst,
                            const float* __restrict__ h, const float* __restrict__ dinv,
                            float* __restrict__ agg, int total)
{
    int idx = blockIdx.x * blockDim.x + threadIdx.x;
    if (idx >= total) return;
    int e = idx >> 3;                         // edge
    int j = idx & 7;                          // feature
    int s = src[e], d = dst[e];
    fadd_atomic(&agg[(size_t)d * HID + j], dinv[s] * dinv[d] * h[(size_t)s * HID + j]);
}

__global__ void k_bias_relu(float* __restrict__ agg, const float* __restrict__ b, int total) {
    int idx = blockIdx.x * blockDim.x + threadIdx.x;
    if (idx >= total) return;
    agg[idx] = fmaxf(agg[idx] + b[idx & 7], 0.0f);
}

// ---------------- layer-2 linear (8 -> 40) + bias + log_softmax, fused per node ----------------
// Uses (A_hat @ h1) @ W2 == A_hat @ (h1 @ W2) to keep edge traffic at 8 dims.
__global__ void k_out(const float* __restrict__ agg2, const float* __restrict__ W2,
                      const float* __restrict__ b2, float* __restrict__ out, int n)
{
    int node = blockIdx.x * blockDim.x + threadIdx.x;
    if (node >= n) return;

    float a[HID];
#pragma unroll
    for (int k = 0; k < HID; ++k) a[k] = agg2[(size_t)node * HID + k];

    float logits[OUT_DIM];
    float mx = -INFINITY;
#pragma unroll
    for (int j = 0; j < OUT_DIM; ++j) {
        float s = b2[j];
#pragma unroll
        for (int k = 0; k < HID; ++k) s = fmaf(a[k], W2[k * OUT_DIM + j], s);
        logits[j] = s;
        mx = fmaxf(mx, s);
    }
    float se = 0.0f;
#pragma unroll
    for (int j = 0; j < OUT_DIM; ++j) se += expf(logits[j] - mx);
    float lse = mx + logf(se);

    size_t base = (size_t)node * OUT_DIM;
    size_t off2 = (size_t)n * OUT_DIM;
#pragma unroll
    for (int j = 0; j < OUT_DIM; ++j) {
        out[base + j]        = logits[j] - lse;  // log_softmax
        out[off2 + base + j] = logits[j];        // raw logits
    }
}

// ---------------- host-side launcher ----------------
extern "C" void kernel_launch(void* const* d_in, const int* in_sizes, int n_in,
                              void* d_out, int out_size, void* d_ws, size_t ws_size,
                              hipStream_t stream)
{
    const float* x  = (const float*)d_in[0];
    const int*   ei = (const int*)  d_in[1];
    const float* W1 = (const float*)d_in[2];
    const float* b1 = (const float*)d_in[3];
    const float* W2 = (const float*)d_in[4];
    const float* b2 = (const float*)d_in[5];
    float* out = (float*)d_out;

    const int n = in_sizes[0] / IN_DIM;       // 100000
    const int E = in_sizes[1] / 2;            // 3200000
    const int* src = ei;
    const int* dst = ei + E;

    // workspace layout (floats): dinv[n] | h1lin[n*8] | agg1[n*8] | agg2[n*8]
    float* dinv  = (float*)d_ws;
    float* h1lin = dinv  + n;
    float* agg1  = h1lin + (size_t)n * HID;
    float* agg2  = agg1  + (size_t)n * HID;

    const int B = 256;
    auto cdiv = [](long a, long b) { return (int)((a + b - 1) / b); };

    // normalization: deg = 1 + incoming-count, then rsqrt in place
    k_deg_init <<<cdiv(n, B), B, 0, stream>>>(dinv, n);
    k_deg_count<<<cdiv(E, B), B, 0, stream>>>(dst, dinv, E);
    k_deg_rsqrt<<<cdiv(n, B), B, 0, stream>>>(dinv, n);

    // layer-1 linear via WMMA (one 16-row tile per wave, 8 waves per block)
    const int ntiles = cdiv(n, 16);
    k_gemm1_wmma<<<cdiv(ntiles, 8), 256, 0, stream>>>(x, W1, h1lin, n);

    // layer-1 propagate + bias + relu  -> agg1 becomes h1
    k_agg_init <<<cdiv((long)n * HID, B), B, 0, stream>>>(h1lin, dinv, agg1, n * HID);
    k_agg_edges<<<cdiv((long)E * HID, B), B, 0, stream>>>(src, dst, h1lin, dinv, agg1, E * HID);
    k_bias_relu<<<cdiv((long)n * HID, B), B, 0, stream>>>(agg1, b1, n * HID);

    // layer-2 propagate first (linearity), then fused 8->40 GEMM + log_softmax
    k_agg_init <<<cdiv((long)n * HID, B), B, 0, stream>>>(agg1, dinv, agg2, n * HID);
    k_agg_edges<<<cdiv((long)E * HID, B), B, 0, stream>>>(src, dst, agg1, dinv, agg2, E * HID);
    k_out      <<<cdiv(n, B), B, 0, stream>>>(agg2, W2, b2, out, n);
}